// renderModule_57389353009223
// MI455X (gfx1250) — compile-verified
//
#include <hip/hip_runtime.h>
#include <math.h>

// ---------------------------------------------------------------------------
// Point-splat renderer for MI455X (gfx1250, wave32).
//   pass 1: zero float4 accumulator (accum.xyz = w*feat, accum.w = wsum)
//   pass 2: WMMA-based projection (V_WMMA_F32_16X16X4_F32: D = Rt(16x4) x P(4x16))
//           + per-lane atomic float scatter into L2-resident 16MB accumulator
//   pass 3: normalize against environment background
//
// Data-movement policy: point stream (96 MB) and environment (12 MB) are
// read-once -> non-temporal loads; output (12 MB) write-once -> non-temporal
// stores. The 16 MB accumulator stays regular-temporal so its lines remain
// resident in the 192 MB L2 for the 16M f32 atomics.
// ---------------------------------------------------------------------------

#define IMG_H 1024
#define IMG_W 1024
#define EPSV 1e-6f
#define BGW  1e-4f

typedef __attribute__((ext_vector_type(2))) float v2f;
typedef __attribute__((ext_vector_type(8))) float v8f;

__global__ __launch_bounds__(256) void zero_acc(float4* __restrict__ acc, int n) {
    int i = blockIdx.x * blockDim.x + threadIdx.x;
    int stride = gridDim.x * blockDim.x;
    for (; i < n; i += stride) acc[i] = make_float4(0.f, 0.f, 0.f, 0.f);
}

// WMMA splat: each wave processes 32 points per iteration via two
// v_wmma_f32_16x16x4_f32 ops (16 points each), then every lane owns one point
// for the projection math + 4 atomic adds.
__global__ __launch_bounds__(256) void splat_wmma(const float* __restrict__ pts,
                                                  const float* __restrict__ cam,
                                                  float* __restrict__ acc,
                                                  unsigned nGroups) {
    const unsigned lane = threadIdx.x & 31u;
    const unsigned m    = lane & 15u;          // matrix row (A) / point-in-group (B cols)
    const bool     hiK  = (lane >= 16u);       // lanes 16..31 hold K = {2,3}
    const unsigned off  = hiK ? 2u : 0u;       // load (z,f0) or (x,y) pair

    // A matrix (16x4 f32): rows 0..2 = Rt rows, rows 3..15 = 0.
    // lane L: VGPR0 = A[m][hiK?2:0], VGPR1 = A[m][hiK?3:1]
    v2f a;
    {
        const unsigned k0 = hiK ? 2u : 0u;
        a.x = (m < 3u) ? cam[m * 4u + k0]      : 0.f;
        a.y = (m < 3u) ? cam[m * 4u + k0 + 1u] : 0.f;
    }
    const float fx = cam[12], fy = cam[13], cx = cam[14], cy = cam[15];

    const unsigned wave   = (blockIdx.x * blockDim.x + threadIdx.x) >> 5;
    const unsigned nWaves = (gridDim.x * blockDim.x) >> 5;

    for (unsigned g = wave; g < nGroups; g += nWaves) {   // wave-uniform loop
        const unsigned p0 = g * 32u;
        const unsigned pl = p0 + m;          // lo group: points p0 .. p0+15
        const unsigned ph = p0 + 16u + m;    // hi group: points p0+16 .. p0+31
        const unsigned p  = p0 + lane;       // point this lane owns for scatter

        // Issue ALL loads for this iteration up front (overlap with WMMA+div).
        // (pl*6+off)*4 bytes = 24*pl {+8} -> always 8B aligned: single b64 loads.
        const v2f tlo = __builtin_nontemporal_load(
            (const v2f*)(pts + (size_t)pl * 6u + off));
        const v2f thi = __builtin_nontemporal_load(
            (const v2f*)(pts + (size_t)ph * 6u + off));
        const float f0 = __builtin_nontemporal_load(pts + (size_t)p * 6u + 3u);
        const float f1 = __builtin_nontemporal_load(pts + (size_t)p * 6u + 4u);
        const float f2 = __builtin_nontemporal_load(pts + (size_t)p * 6u + 5u);

        // B matrix (4x16 f32), homogeneous coords [x;y;z;1]:
        //   VGPR0: lanes0-15 = row K=0 (x), lanes16-31 = row K=2 (z)
        //   VGPR1: lanes0-15 = row K=1 (y), lanes16-31 = row K=3 (1.0)
        v2f blo, bhi;
        blo.x = tlo.x;  blo.y = hiK ? 1.0f : tlo.y;
        bhi.x = thi.x;  bhi.y = hiK ? 1.0f : thi.y;

        v8f c0 = {};
        // D[m][n] = sum_k A[m][k]*B[k][n]  ->  rows 0/1/2 = x_cam/y_cam/z_cam
        v8f dlo = __builtin_amdgcn_wmma_f32_16x16x4_f32(
            false, a, false, blo, (short)0, c0, false, false);
        v8f dhi = __builtin_amdgcn_wmma_f32_16x16x4_f32(
            false, a, false, bhi, (short)0, c0, false, false);

        // Redistribute: lane L takes point p0+L (x,y,z live in D VGPR0/1/2,
        // lane L&15 of the matching group).
        const int sel = (int)m;
        float X, Y, Z;
        {
            float xl = __shfl(dlo[0], sel, 32), xh = __shfl(dhi[0], sel, 32);
            float yl = __shfl(dlo[1], sel, 32), yh = __shfl(dhi[1], sel, 32);
            float zl = __shfl(dlo[2], sel, 32), zh = __shfl(dhi[2], sel, 32);
            X = hiK ? xh : xl;
            Y = hiK ? yh : yl;
            Z = hiK ? zh : zl;
        }

        const float inv_z = 1.0f / fmaxf(Z, EPSV);
        const float u = fx * X * inv_z + cx;
        const float v = fy * Y * inv_z + cy;
        const bool valid = (Z > EPSV) & (u >= 0.f) & (u < (float)IMG_W)
                                      & (v >= 0.f) & (v < (float)IMG_H);
        if (valid) {
            const int ui = (int)u;           // u >= 0 -> trunc == floor
            const int vi = (int)v;
            const int idx = vi * IMG_W + ui;
            float* cell = acc + (size_t)idx * 4u;   // float4 cell: one cacheline
            atomicAdd(cell + 0, inv_z * f0);
            atomicAdd(cell + 1, inv_z * f1);
            atomicAdd(cell + 2, inv_z * f2);
            atomicAdd(cell + 3, inv_z);
        }
    }
}

// Scalar tail for N not divisible by 32 (defensive; N=4M is divisible).
__global__ void splat_tail(const float* __restrict__ pts,
                           const float* __restrict__ cam,
                           float* __restrict__ acc,
                           unsigned pStart, unsigned nPts) {
    unsigned t = blockIdx.x * blockDim.x + threadIdx.x;
    if (t >= nPts) return;
    const unsigned p = pStart + t;
    const float r00 = cam[0], r01 = cam[1], r02 = cam[2],  t0 = cam[3];
    const float r10 = cam[4], r11 = cam[5], r12 = cam[6],  t1 = cam[7];
    const float r20 = cam[8], r21 = cam[9], r22 = cam[10], t2 = cam[11];
    const float fx = cam[12], fy = cam[13], cx = cam[14], cy = cam[15];
    const float x = pts[p*6+0], y = pts[p*6+1], z = pts[p*6+2];
    const float X = r00*x + r01*y + r02*z + t0;
    const float Y = r10*x + r11*y + r12*z + t1;
    const float Z = r20*x + r21*y + r22*z + t2;
    const float inv_z = 1.0f / fmaxf(Z, EPSV);
    const float u = fx * X * inv_z + cx;
    const float v = fy * Y * inv_z + cy;
    if ((Z > EPSV) & (u >= 0.f) & (u < (float)IMG_W) & (v >= 0.f) & (v < (float)IMG_H)) {
        const int idx = (int)v * IMG_W + (int)u;
        float* cell = acc + (size_t)idx * 4u;
        atomicAdd(cell + 0, inv_z * pts[p*6+3]);
        atomicAdd(cell + 1, inv_z * pts[p*6+4]);
        atomicAdd(cell + 2, inv_z * pts[p*6+5]);
        atomicAdd(cell + 3, inv_z);
    }
}

__global__ __launch_bounds__(256) void finalize(const float4* __restrict__ acc,
                                                const float* __restrict__ env,
                                                float* __restrict__ out, int n) {
    int i = blockIdx.x * blockDim.x + threadIdx.x;
    int stride = gridDim.x * blockDim.x;
    for (; i < n; i += stride) {
        const float4 a = acc[i];
        const float inv = 1.0f / (a.w + BGW);
        const float e0 = __builtin_nontemporal_load(env + (size_t)i * 3u + 0u);
        const float e1 = __builtin_nontemporal_load(env + (size_t)i * 3u + 1u);
        const float e2 = __builtin_nontemporal_load(env + (size_t)i * 3u + 2u);
        __builtin_nontemporal_store((a.x + BGW * e0) * inv, out + (size_t)i * 3u + 0u);
        __builtin_nontemporal_store((a.y + BGW * e1) * inv, out + (size_t)i * 3u + 1u);
        __builtin_nontemporal_store((a.z + BGW * e2) * inv, out + (size_t)i * 3u + 2u);
    }
}

extern "C" void kernel_launch(void* const* d_in, const int* in_sizes, int n_in,
                              void* d_out, int out_size, void* d_ws, size_t ws_size,
                              hipStream_t stream) {
    // inputs: [0] cam_type (int, ignored; pinhole), [1] camera (16 f32),
    //         [2] points (N*6 f32), [3] environment (H*W*3 f32)
    const float* cam = (const float*)d_in[1];
    const float* pts = (const float*)d_in[2];
    const float* env = (const float*)d_in[3];
    float* out = (float*)d_out;
    float* acc = (float*)d_ws;                    // needs H*W*4*4 = 16 MB
    const unsigned N  = (unsigned)(in_sizes[2] / 6);
    const int HW = IMG_H * IMG_W;

    zero_acc<<<2048, 256, 0, stream>>>((float4*)acc, HW);

    const unsigned nGroups = N / 32u;
    if (nGroups)
        splat_wmma<<<3072, 256, 0, stream>>>(pts, cam, acc, nGroups);
    const unsigned rem = N - nGroups * 32u;
    if (rem)
        splat_tail<<<(rem + 255u) / 256u, 256, 0, stream>>>(pts, cam, acc,
                                                            nGroups * 32u, rem);

    finalize<<<4096, 256, 0, stream>>>((const float4*)acc, env, out, HW);
}